// GraphConv2d_60997125538366
// MI455X (gfx1250) — compile-verified
//
#include <hip/hip_runtime.h>

typedef __attribute__((ext_vector_type(2))) float v2f;
typedef __attribute__((ext_vector_type(4))) float v4f;
typedef __attribute__((ext_vector_type(8))) float v8f;

#define Bq   2
#define Cc   64
#define Nn   65536
#define Kk   16
#define OUTc 64

// ---------------------------------------------------------------------------
// Kernel 1: per-node linear transforms via V_WMMA_F32_16X16X4_F32 (f32 exact)
//   c_buf[b*N+n][o] = sum_i (W[o][i] - W[o][64+i]) * x[b][i][n] + bias[o]
//   y_buf[b*N+n][o] = sum_i  W[o][64+i]            * x[b][i][n]
// Outputs stored node-major ([node][64]) so kernel 2 gathers are contiguous.
// ---------------------------------------------------------------------------
__global__ __launch_bounds__(256) void edge_pre_gemm(
    const float* __restrict__ x, const float* __restrict__ W,
    const float* __restrict__ bias, float* __restrict__ c_buf,
    float* __restrict__ y_buf)
{
  constexpr int NPB = 256;                       // nodes per block
  constexpr int blocksPerBatch = Nn / NPB;
  const int b     = blockIdx.x / blocksPerBatch;
  const int nbase = (blockIdx.x % blocksPerBatch) * NPB;

  const int wave   = threadIdx.x >> 5;
  const int lane   = threadIdx.x & 31;
  const int l16    = lane & 15;
  const int hi     = lane >> 4;                  // half-wave select
  const int khalf  = hi * 2;                     // A/B K offset: 0 or 2
  const int m0     = (wave & 3) * 16;            // output-channel tile
  const int matsel = wave >> 2;                  // 0 = Wc (+bias), 1 = Wn

  // Preload A operands (weights) for all 16 K-steps: 32 VGPRs, reused
  // across every node tile this block processes.
  float a_lo[16], a_hi[16];
  const int row = m0 + l16;                      // A-matrix M = lane
  #pragma unroll
  for (int kk = 0; kk < 16; ++kk) {
    const int ci  = kk * 4 + khalf;
    const float w1a = W[row * 128 + ci];
    const float w2a = W[row * 128 + 64 + ci];
    const float w1b = W[row * 128 + ci + 1];
    const float w2b = W[row * 128 + 64 + ci + 1];
    a_lo[kk] = matsel ? w2a : (w1a - w2a);
    a_hi[kk] = matsel ? w2b : (w1b - w2b);
  }

  // Accumulator seed: bias for the center matrix, zero for the neighbor one.
  // D layout: VGPR v -> M = m0 + v + 8*hi, lane(&15) -> N (node).
  v8f cinit;
  #pragma unroll
  for (int v = 0; v < 8; ++v)
    cinit[v] = matsel ? 0.0f : bias[m0 + v + hi * 8];

  const float* xb  = x + (size_t)b * Cc * Nn;
  float*       dst = (matsel ? y_buf : c_buf) + (size_t)b * Nn * OUTc;

  for (int t = 0; t < NPB / 16; ++t) {
    const int node = nbase + t * 16 + l16;       // B-matrix N = lane
    v8f acc = cinit;
    #pragma unroll
    for (int kk = 0; kk < 16; ++kk) {
      const int ch = kk * 4 + khalf;
      v2f bm;
      bm.x = xb[(size_t)ch * Nn + node];         // K = khalf
      bm.y = xb[(size_t)(ch + 1) * Nn + node];   // K = khalf + 1
      v2f am;
      am.x = a_lo[kk];
      am.y = a_hi[kk];
      acc = __builtin_amdgcn_wmma_f32_16x16x4_f32(
          false, am, false, bm, (short)0, acc, false, false);
    }
    // Each lane stores 8 consecutive channels: two b128 stores.
    float* p = dst + (size_t)node * OUTc + m0 + hi * 8;
    *(v4f*)(p)     = *(const v4f*)&acc;
    *(v4f*)(p + 4) = *((const v4f*)&acc + 1);
  }
}

// ---------------------------------------------------------------------------
// Kernel 2: neighbor gather + fused (add, relu, max-over-K).
// One wave per node; lane l handles channels 2l, 2l+1 (coalesced 256B gather).
// max_k relu(c + y_j) == max(0, max_k (c + y_j)) -> init acc = 0.
// ---------------------------------------------------------------------------
__global__ __launch_bounds__(256) void edge_gather_max(
    const long long* __restrict__ eidx, const float* __restrict__ c_buf,
    const float* __restrict__ y_buf, float* __restrict__ out)
{
  const int wave = threadIdx.x >> 5;
  const int lane = threadIdx.x & 31;
  // Force the node id scalar so the int64 index reads become SMEM loads.
  const int gnode = __builtin_amdgcn_readfirstlane((int)blockIdx.x * 8 + wave);
  const int b = gnode >> 16;                     // gnode / N
  const int n = gnode & (Nn - 1);                // gnode % N

  const v2f cv = *(const v2f*)(c_buf + (size_t)gnode * OUTc + lane * 2);
  const float*     yb = y_buf + (size_t)b * Nn * OUTc;
  const long long* ip = eidx + (size_t)gnode * Kk;   // edge_index[0][b][n][:]

  v2f acc;
  acc.x = 0.0f;
  acc.y = 0.0f;
  #pragma unroll
  for (int k = 0; k < Kk; ++k) {
    const long long j = ip[k];
    const v2f yv = *(const v2f*)(yb + (size_t)j * OUTc + lane * 2);
    acc.x = fmaxf(acc.x, cv.x + yv.x);
    acc.y = fmaxf(acc.y, cv.y + yv.y);
  }
  // out[b][o][n][1], o = 2*lane, 2*lane+1 (channel-major like the reference)
  float* op = out + ((size_t)b * OUTc + (size_t)(lane * 2)) * Nn + n;
  op[0]  = acc.x;
  op[Nn] = acc.y;
}

extern "C" void kernel_launch(void* const* d_in, const int* in_sizes, int n_in,
                              void* d_out, int out_size, void* d_ws, size_t ws_size,
                              hipStream_t stream) {
  const float*     x    = (const float*)d_in[0];
  const long long* eidx = (const long long*)d_in[1];   // int64 edge_index
  const float*     W    = (const float*)d_in[2];
  const float*     bias = (const float*)d_in[3];
  float*           out  = (float*)d_out;

  float* c_buf = (float*)d_ws;                                  // 32 MB
  float* y_buf = c_buf + (size_t)Bq * Nn * OUTc;                // 32 MB

  edge_pre_gemm<<<dim3(Bq * (Nn / 256)), dim3(256), 0, stream>>>(
      x, W, bias, c_buf, y_buf);
  edge_gather_max<<<dim3(Bq * Nn / 8), dim3(256), 0, stream>>>(
      eidx, c_buf, y_buf, out);
}